// Net_5437428596910
// MI455X (gfx1250) — compile-verified
//
#include <hip/hip_runtime.h>

#define H   10
#define G4  40   // 4*H gate width

typedef float v2f __attribute__((ext_vector_type(2)));
typedef float v8f __attribute__((ext_vector_type(8)));

__device__ __forceinline__ float fast_sig(float x) {
    // sigmoid(x) = 1 / (1 + 2^(-x*log2(e))); saturates cleanly via exp2/rcp
    float e = __builtin_amdgcn_exp2f(-x * 1.4426950408889634f);
    return __builtin_amdgcn_rcpf(1.0f + e);
}

// ---------------------------------------------------------------------------
// Kernel 1: x_proj[t][j] = sum_k x[t][k] * w_ih[j][k] + b_ih[j] + b_hh[j]
// One wave per 16-row tile of x; 3 N-tiles (N=48 padded over 40), K padded
// 10 -> 12 as three V_WMMA_F32_16X16X4_F32 accumulations.
// ---------------------------------------------------------------------------
__global__ __launch_bounds__(32) void lstm_xproj_wmma(
    const float* __restrict__ x, const float* __restrict__ w_ih,
    const float* __restrict__ b_ih, const float* __restrict__ b_hh,
    float* __restrict__ xp, int T) {
    const int lane = threadIdx.x;      // 0..31
    const int m0   = blockIdx.x * 16;
    const int half = lane >> 4;        // 0: lanes 0-15, 1: lanes 16-31
    const int l16  = lane & 15;

    // A fragment (16x4 f32): vgpr0 = K{0,2}, vgpr1 = K{1,3} split by lane half
    const int mrow = m0 + l16;
    v2f a[3];
#pragma unroll
    for (int kc = 0; kc < 3; ++kc) {
        int kb  = kc * 4 + half * 2;
        float a0 = x[mrow * H + (kb     < H ? kb     : 0)];
        float a1 = x[mrow * H + (kb + 1 < H ? kb + 1 : 0)];
        a[kc].x = (kb     < H) ? a0 : 0.0f;
        a[kc].y = (kb + 1 < H) ? a1 : 0.0f;
    }

#pragma unroll
    for (int nt = 0; nt < 3; ++nt) {
        const int col  = nt * 16 + l16;
        const int colc = (col < G4) ? col : 0;
        v8f acc = {};
#pragma unroll
        for (int kc = 0; kc < 3; ++kc) {
            int kb  = kc * 4 + half * 2;
            // B fragment (4x16 f32): B[k][n] = w_ih[n*H + k]
            float b0 = w_ih[colc * H + (kb     < H ? kb     : 0)];
            float b1 = w_ih[colc * H + (kb + 1 < H ? kb + 1 : 0)];
            v2f b;
            b.x = (col < G4 && kb     < H) ? b0 : 0.0f;
            b.y = (col < G4 && kb + 1 < H) ? b1 : 0.0f;
            acc = __builtin_amdgcn_wmma_f32_16x16x4_f32(
                false, a[kc], false, b, (short)0, acc, false, false);
        }
        float bias = (col < G4) ? (b_ih[colc] + b_hh[colc]) : 0.0f;
        if (col < G4) {
#pragma unroll
            for (int r = 0; r < 8; ++r) {
                int m = m0 + r + half * 8;   // C/D layout: vgpr r = rows r, r+8
                xp[m * G4 + col] = acc[r] + bias;
            }
        }
    }
}

// ---------------------------------------------------------------------------
// Kernel 2: the 65536-step serial recurrence on a single wave32.
// Lane j owns gate rows j and j+20 of w_hh (gate order i,f,g,o):
//   lanes 0-9 : gA = i[j]  (sigmoid),  gB = g[j] (tanh)
//   lanes10-19: gA = f[j-10](sigmoid), gB = o[j-10] (sigmoid)
// c lives on lanes 0-9; f/o arrive via __shfl from lane+10; h is broadcast
// back to all lanes with v_readlane (SGPR operands for next step's FMAs).
// ---------------------------------------------------------------------------
__global__ __launch_bounds__(32) void lstm_scan(
    const float* __restrict__ xp, const float* __restrict__ w_hh,
    const float* __restrict__ h0, const float* __restrict__ c0,
    float* __restrict__ out, int T, int outRows) {
    const int lane = threadIdx.x;
    const int g1   = lane;
    const int g2   = lane + 20;
    const int g1c  = (g1 < G4) ? g1 : 0;
    const int g2c  = (g2 < G4) ? g2 : 0;

    float wA[H], wB[H];
#pragma unroll
    for (int k = 0; k < H; ++k) {
        wA[k] = w_hh[g1c * H + k];
        wB[k] = w_hh[g2c * H + k];
    }

    // actB nonlinearity: lanes 0-9 tanh(x)=2*sig(2x)-1, lanes 10+ sigmoid(x)
    const bool tB = (lane < H);
    const float sB = tB ? 2.0f : 1.0f;
    const float mB = tB ? 2.0f : 1.0f;
    const float aB = tB ? -1.0f : 0.0f;

    float hbc[H];
#pragma unroll
    for (int k = 0; k < H; ++k) hbc[k] = h0[k];
    float c = c0[(lane < H) ? lane : 0];

    const int firstOut = T - outRows;
    float xpA = xp[g1c];
    float xpB = xp[g2c];

    for (int t = 0; t < T; ++t) {
        // register-level prefetch of next row + L0 prefetch ~24 rows ahead
        int tn = (t + 1 < T) ? t + 1 : t;
        float nxA = xp[tn * G4 + g1c];
        float nxB = xp[tn * G4 + g2c];
        int tp = t + 24; tp = (tp < T) ? tp : T - 1;
        __builtin_prefetch(&xp[tp * G4 + g1c], 0, 0);

        // two length-10 dot products, 4 partial sums to shorten the chain
        float pa0 = xpA, pa1 = 0.f, pa2 = 0.f, pa3 = 0.f;
        float pb0 = xpB, pb1 = 0.f, pb2 = 0.f, pb3 = 0.f;
#pragma unroll
        for (int k = 0; k < H; k += 4) {
            pa0 = __builtin_fmaf(hbc[k],     wA[k],     pa0);
            pa1 = __builtin_fmaf(hbc[k + 1], wA[k + 1], pa1);
            pb0 = __builtin_fmaf(hbc[k],     wB[k],     pb0);
            pb1 = __builtin_fmaf(hbc[k + 1], wB[k + 1], pb1);
            if (k + 2 < H) {
                pa2 = __builtin_fmaf(hbc[k + 2], wA[k + 2], pa2);
                pa3 = __builtin_fmaf(hbc[k + 3], wA[k + 3], pa3);
                pb2 = __builtin_fmaf(hbc[k + 2], wB[k + 2], pb2);
                pb3 = __builtin_fmaf(hbc[k + 3], wB[k + 3], pb3);
            }
        }
        float gA = (pa0 + pa1) + (pa2 + pa3);
        float gB = (pb0 + pb1) + (pb2 + pb3);

        float actA = fast_sig(gA);                              // i or f
        float actB = __builtin_fmaf(fast_sig(gB * sB), mB, aB); // g or o

        int src = (lane + 10) & 31;
        float f_sh = __shfl(actA, src);   // lanes 0-9 get f[j]
        float o_sh = __shfl(actB, src);   // lanes 0-9 get o[j]

        c = __builtin_fmaf(f_sh, c, actA * actB);               // f*c + i*g
        float tc = __builtin_fmaf(fast_sig(2.0f * c), 2.0f, -1.0f); // tanh(c)
        float h  = o_sh * tc;

        if (t >= firstOut && lane < H)
            out[(t - firstOut) * H + lane] = h;

        // broadcast h[0..9] to all lanes as uniform values
#pragma unroll
        for (int k = 0; k < H; ++k)
            hbc[k] = __int_as_float(
                __builtin_amdgcn_readlane(__float_as_int(h), k));

        xpA = nxA; xpB = nxB;
    }
}

extern "C" void kernel_launch(void* const* d_in, const int* in_sizes, int n_in,
                              void* d_out, int out_size, void* d_ws, size_t ws_size,
                              hipStream_t stream) {
    const float* x    = (const float*)d_in[0];
    const float* w_ih = (const float*)d_in[1];
    const float* w_hh = (const float*)d_in[2];
    const float* b_ih = (const float*)d_in[3];
    const float* b_hh = (const float*)d_in[4];
    const float* h0   = (const float*)d_in[5];
    const float* c0   = (const float*)d_in[6];
    float* out = (float*)d_out;
    float* xp  = (float*)d_ws;              // T*40 floats (~10.5 MB)

    const int T       = in_sizes[0] / H;    // 65536
    const int outRows = out_size / H;       // 64

    lstm_xproj_wmma<<<T / 16, 32, 0, stream>>>(x, w_ih, b_ih, b_hh, xp, T);
    lstm_scan<<<1, 32, 0, stream>>>(xp, w_hh, h0, c0, out, T, outRows);
}